// StructureLoss_46866683134317
// MI455X (gfx1250) — compile-verified
//
#include <hip/hip_runtime.h>
#include <hip/hip_bf16.h>

// Problem constants from the reference
#define N_   8
#define C_   256
#define H_   96
#define W_   96
#define A_   4096
#define HW_  (H_ * W_)
#define ANCHORS_ (N_ * A_)
#define WAVES_PER_BLOCK_ 8

typedef float v2f __attribute__((ext_vector_type(2)));
typedef float v8f __attribute__((ext_vector_type(8)));

// Reflect-pad coordinate map: v in [-1, H] -> source row/col in [0, H-1]
__device__ __forceinline__ int reflect_coord(int v, int lim) {
    int q = v < 0 ? -v : v;
    q = q > (lim - 1) ? (2 * (lim - 1) - q) : q;
    return q;
}

// One wave per anchor. Builds the 16x16 Gram matrix G = S * S^T for the
// 9x256 structure matrix S via 64 x V_WMMA_F32_16X16X4_F32, for both
// feature maps, then emits sum_k |sv1_k - sv2_k| per anchor.
//
// Padding rows 9..15 of S are NOT zeroed: they only influence G rows/cols
// 9..15, which are never read. Padding lanes gather from off_sp=0, so the
// inner loop runs with full EXEC and no predication.
__global__ __launch_bounds__(WAVES_PER_BLOCK_ * 32)
void structure_gram_kernel(const float* __restrict__ feat1,
                           const float* __restrict__ feat2,
                           const int*   __restrict__ index,
                           float*       __restrict__ partial) {
    __shared__ float lds[WAVES_PER_BLOCK_ * 256];

    const int lane   = threadIdx.x & 31;
    const int wave   = threadIdx.x >> 5;
    const int anchor = blockIdx.x * WAVES_PER_BLOCK_ + wave;   // grid sized exactly
    const int n = anchor / A_;
    const int a = anchor % A_;

    // This lane's matrix row (position index M); rows 9..15 are don't-care.
    const int M = lane & 15;

    int off_sp = 0;  // spatial element offset qx*W + qy after reflect padding
    if (M < 9) {
        const size_t ib = ((size_t)(n * A_ + a) * 9 + M) * 2;
        const int vx = index[ib + 0];
        const int vy = index[ib + 1];
        off_sp = reflect_coord(vx, H_) * W_ + reflect_coord(vy, W_);
    }

    // fp32 A-fragment: lanes 0..15 hold S[M][K+0..1], lanes 16..31 hold S[M][K+2..3]
    const int cbase = (lane >= 16) ? 2 : 0;
    float* slot = &lds[wave * 256];

    float svf[2];
    #pragma unroll 1
    for (int f = 0; f < 2; ++f) {
        const float* base = ((f == 0) ? feat1 : feat2) + (size_t)n * C_ * HW_;
        const float* p = base + (size_t)cbase * HW_ + off_sp;

        v8f acc = {};  // G accumulator, 16x16 f32
        #pragma unroll 4
        for (int i = 0; i < 64; ++i) {
            const float a0 = p[0];
            const float a1 = p[HW_];
            p += 4 * HW_;

            // B = S^T fragment from the A fragment via a half-wave swap.
            const float s0 = __shfl_xor(a0, 16, 32);
            const float s1 = __shfl_xor(a1, 16, 32);
            const float b0 = (lane < 16) ? a0 : s1;  // rows K+0 / K+1
            const float b1 = (lane < 16) ? s0 : a1;  // rows K+2 / K+3

            v2f av; av.x = a0; av.y = a1;
            v2f bv; bv.x = b0; bv.y = b1;
            acc = __builtin_amdgcn_wmma_f32_16x16x4_f32(
                /*neg_a=*/false, av, /*neg_b=*/false, bv,
                /*c_mod=*/(short)0, acc, /*reuse_a=*/false, /*reuse_b=*/false);
        }

        // Spill the Gram tile: acc[r] holds G[r + (lane<16?0:8)][lane&15]
        const int rbase = (lane >= 16) ? 8 : 0;
        const int col   = lane & 15;
        #pragma unroll
        for (int r = 0; r < 8; ++r)
            slot[(rbase + r) * 16 + col] = acc[r];
        __syncthreads();

        // Lanes 0..7 compute cosine similarity for neighbor k = lane+1
        float svv = 0.0f;
        if (lane < 8) {
            const int   k   = lane + 1;
            const float g00 = slot[0];            // |center|^2
            const float dot = slot[k * 16 + 0];   // rnd_k . center
            const float nsq = slot[k * 16 + k];   // |rnd_k|^2
            svv = dot * rsqrtf(nsq * g00);
        }
        svf[f] = svv;
        __syncthreads();
    }

    // Per-anchor sum of |sv1 - sv2| over the 8 neighbors (lanes 0..7)
    float d = (lane < 8) ? fabsf(svf[0] - svf[1]) : 0.0f;
    d += __shfl_xor(d, 1, 32);
    d += __shfl_xor(d, 2, 32);
    d += __shfl_xor(d, 4, 32);
    if (lane == 0) partial[anchor] = d;
}

// Deterministic single-block tree reduction -> mean
__global__ __launch_bounds__(256)
void reduce_mean_kernel(const float* __restrict__ partial,
                        float* __restrict__ out,
                        int n, float scale) {
    __shared__ float s[256];
    const int t = threadIdx.x;
    float sum = 0.0f;
    for (int i = t; i < n; i += 256) sum += partial[i];
    s[t] = sum;
    __syncthreads();
    #pragma unroll
    for (int off = 128; off > 0; off >>= 1) {
        if (t < off) s[t] += s[t + off];
        __syncthreads();
    }
    if (t == 0) out[0] = s[0] * scale;
}

extern "C" void kernel_launch(void* const* d_in, const int* in_sizes, int n_in,
                              void* d_out, int out_size, void* d_ws, size_t ws_size,
                              hipStream_t stream) {
    const float* feat1 = (const float*)d_in[0];
    const float* feat2 = (const float*)d_in[1];
    const int*   index = (const int*)d_in[2];
    float* out     = (float*)d_out;
    float* partial = (float*)d_ws;   // 32768 floats = 128 KB scratch

    (void)in_sizes; (void)n_in; (void)out_size; (void)ws_size;

    structure_gram_kernel<<<ANCHORS_ / WAVES_PER_BLOCK_, WAVES_PER_BLOCK_ * 32, 0, stream>>>(
        feat1, feat2, index, partial);

    const float scale = 1.0f / (float)(ANCHORS_ * 8);
    reduce_mean_kernel<<<1, 256, 0, stream>>>(partial, out, ANCHORS_, scale);
}